// SMRNN_30451318128726
// MI455X (gfx1250) — compile-verified
//
#include <hip/hip_runtime.h>
#include <math.h>

typedef __attribute__((ext_vector_type(2))) float v2f;
typedef __attribute__((ext_vector_type(8))) float v8f;

// ---------------------------------------------------------------------------
// concat: inpt[448] = [x(256), prev_output(64), state(128)]
// ---------------------------------------------------------------------------
__global__ void concat_kernel(const float* __restrict__ x,
                              const float* __restrict__ po,
                              const float* __restrict__ st,
                              float* __restrict__ out) {
    int i = blockIdx.x * blockDim.x + threadIdx.x;
    if (i < 448) {
        float v;
        if (i < 256)      v = x[i];
        else if (i < 320) v = po[i - 256];
        else              v = st[i - 320];
        out[i] = v;
    }
}

// ---------------------------------------------------------------------------
// y[M] = act( (W [+ dW]) @ x + bscale*b [+ b2] )
// Row-major W (M x K). K % 4 == 0, M % 16 == 0, K <= 512.
// One wave32 computes 16 output rows via V_WMMA_F32_16X16X4_F32:
//   A fragment (16x4 f32): lane l holds W[row_base+(l&15), k0+(l>>4)*2 .. +1]
//                          -> one contiguous b64 global load per lane per chunk
//   B fragment (4x16 f32): x[k] broadcast across all 16 columns
//                          -> lane l needs x[k0+(l>>4)*2 .. +1] from LDS
//   C/D (16x16 f32): column n lives in lanes n (rows 0-7, c[0..7]) and
//                    16+n (rows 8-15); lanes 0 and 16 write results.
// act: 0=none, 1=leaky(0.01), 2=tanh*5, 3=tanh*2
// ---------------------------------------------------------------------------
__global__ void __launch_bounds__(256)
wmma_matvec_kernel(const float* __restrict__ W,
                   const float* __restrict__ dW,   // nullptr if unused
                   const float* __restrict__ x,
                   const float* __restrict__ b,    // nullptr if unused
                   float bscale,
                   const float* __restrict__ b2,   // nullptr if unused
                   float* __restrict__ y,
                   int M, int K, int act) {
    __shared__ float sx[512];
    for (int i = threadIdx.x; i < K; i += blockDim.x) sx[i] = x[i];
    __syncthreads();

    const int lane = threadIdx.x & 31;
    const int wave = threadIdx.x >> 5;                 // 0..7 (wave32)
    const long long tile = (long long)blockIdx.x * (blockDim.x >> 5) + wave;
    const long long row_base = tile * 16;
    if (row_base >= M) return;                         // wave-uniform: EXEC stays full

    const int rm   = lane & 15;                        // row of A fragment
    const int ksub = (lane >> 4) * 2;                  // 0 or 2: column sub-offset

    const float* __restrict__ wrow = W + (row_base + rm) * (long long)K + ksub;
    const float* __restrict__ drow =
        dW ? dW + (row_base + rm) * (long long)K + ksub : nullptr;

    v8f c = {};
    if (drow) {
        for (int k0 = 0; k0 < K; k0 += 4) {
            v2f a = *(const v2f*)(wrow + k0);
            v2f d = *(const v2f*)(drow + k0);
            a.x += d.x; a.y += d.y;
            v2f bb; bb.x = sx[k0 + ksub]; bb.y = sx[k0 + ksub + 1];
            c = __builtin_amdgcn_wmma_f32_16x16x4_f32(false, a, false, bb,
                                                      (short)0, c, false, false);
        }
    } else {
        for (int k0 = 0; k0 < K; k0 += 4) {
            v2f a = *(const v2f*)(wrow + k0);
            v2f bb; bb.x = sx[k0 + ksub]; bb.y = sx[k0 + ksub + 1];
            c = __builtin_amdgcn_wmma_f32_16x16x4_f32(false, a, false, bb,
                                                      (short)0, c, false, false);
        }
    }

    // Lanes 0 and 16 hold column 0 of D: rows 0-7 / 8-15 respectively.
    if ((lane & 15) == 0) {
        const long long m0 = row_base + (lane >> 4) * 8;
        #pragma unroll
        for (int r = 0; r < 8; ++r) {
            long long m = m0 + r;
            float v = c[r];
            if (b)  v += bscale * b[m];
            if (b2) v += b2[m];
            if (act == 1)      v = (v >= 0.0f) ? v : 0.01f * v;
            else if (act == 2) v = tanhf(v) * 5.0f;
            else if (act == 3) v = tanhf(v) * 2.0f;
            y[m] = v;
        }
    }
}

// ---------------------------------------------------------------------------
// Topologies:
//   T1 = [448, 512, 512, 64]
//   T2 = [448, 448, 512, 512, 64, 734848]
//   T3 = [448, 448, 512, 512, 64, 128]
//   W3TOT = 733184, B3TOT = 1664
// d_in order: 0:x 1:prev 2:state | 3-5:net1_ws 6-8:net1_bs |
//             9-13:net2_ws 14-18:net2_bs | 19-23:net3_ws 24-28:net3_bs
// d_out: [x1(64), new_state(128)]
// ---------------------------------------------------------------------------
extern "C" void kernel_launch(void* const* d_in, const int* in_sizes, int n_in,
                              void* d_out, int out_size, void* d_ws, size_t ws_size,
                              hipStream_t stream) {
    const float* x_in = (const float*)d_in[0];
    const float* prev = (const float*)d_in[1];
    const float* st   = (const float*)d_in[2];

    const float* W1[3]; const float* B1[3];
    const float* W2[5]; const float* B2[5];
    const float* W3[5]; const float* B3[5];
    for (int i = 0; i < 3; ++i) { W1[i] = (const float*)d_in[3 + i];  B1[i] = (const float*)d_in[6 + i]; }
    for (int i = 0; i < 5; ++i) { W2[i] = (const float*)d_in[9 + i];  B2[i] = (const float*)d_in[14 + i]; }
    for (int i = 0; i < 5; ++i) { W3[i] = (const float*)d_in[19 + i]; B3[i] = (const float*)d_in[24 + i]; }

    float* out = (float*)d_out;
    float* ws  = (float*)d_ws;
    float* inpt = ws;            // 448
    float* bufA = ws + 512;      // up to 512
    float* bufB = ws + 1024;     // up to 512
    float* x2   = ws + 2048;     // 734848 = W3TOT + B3TOT
    const long long W3TOT = 733184;
    float* wc = x2;
    float* bc = x2 + W3TOT;

    concat_kernel<<<2, 256, 0, stream>>>(x_in, prev, st, inpt);

    auto mv = [&](const float* W, const float* dW, const float* xin,
                  const float* b, float bscale, const float* b2,
                  float* y, int M, int K, int act) {
        int waves  = (M + 15) / 16;
        int blocks = (waves + 7) / 8;          // 8 wave32s (256 threads) per block
        wmma_matvec_kernel<<<blocks, 256, 0, stream>>>(W, dW, xin, b, bscale, b2,
                                                       y, M, K, act);
    };

    // ---- net2: hypernetwork producing x2 = tanh(mlp(inpt)) * 5 ----
    mv(W2[0], nullptr, inpt, B2[0], 1.0f, nullptr, bufA, 448, 448, 1);
    mv(W2[1], nullptr, bufA, B2[1], 1.0f, nullptr, bufB, 512, 448, 1);
    mv(W2[2], nullptr, bufB, B2[2], 1.0f, nullptr, bufA, 512, 512, 1);
    mv(W2[3], nullptr, bufA, B2[3], 1.0f, nullptr, bufB,  64, 512, 1);
    // The 188 MB layer: tanh*5 fused into the store.
    mv(W2[4], nullptr, bufB, B2[4], 1.0f, nullptr, x2, 734848, 64, 2);

    // ---- net3: modulated MLP, x3_0 = inpt; bias = 2*b3 + bc slice ----
    // wc row offsets: 0, 200704, 430080, 692224, 724992
    // bc offsets:     0, 448, 960, 1472, 1536
    mv(W3[0], wc +      0, inpt, B3[0], 2.0f, bc +    0, bufA, 448, 448, 1);
    mv(W3[1], wc + 200704, bufA, B3[1], 2.0f, bc +  448, bufB, 512, 448, 1);
    mv(W3[2], wc + 430080, bufB, B3[2], 2.0f, bc +  960, bufA, 512, 512, 1);
    mv(W3[3], wc + 692224, bufA, B3[3], 2.0f, bc + 1472, bufB,  64, 512, 1);
    // new_state = tanh(last layer) * 2 -> d_out[64..192)
    mv(W3[4], wc + 724992, bufB, B3[4], 2.0f, bc + 1536, out + 64, 128, 64, 3);

    // ---- net1: plain MLP -> x1 -> d_out[0..64) ----
    mv(W1[0], nullptr, inpt, B1[0], 1.0f, nullptr, bufA, 512, 448, 1);
    mv(W1[1], nullptr, bufA, B1[1], 1.0f, nullptr, bufB, 512, 512, 1);
    mv(W1[2], nullptr, bufB, B1[2], 1.0f, nullptr, out,  64, 512, 0);
}